// PlainVoxels_25675314496067
// MI455X (gfx1250) — compile-verified
//
#include <hip/hip_runtime.h>
#include <hip/hip_bf16.h>

// ---------------------------------------------------------------------------
// PlainVoxels (VolSDF-style hash-grid renderer) for MI455X / gfx1250.
// Single fused kernel:
//   Phase 1 (all 8 waves): 2 lanes per sample (lane n & n+16 share sample n).
//     Each lane gathers 4 of the 8 trilinear corners (7x global_load_b128 per
//     corner), accumulates partial emb[28] + analytic sdf-gradient, then one
//     shfl_xor(16) pair-reduction gives both lanes the full embedding. The SH
//     color projection uses V_WMMA_F32_16X16X4_F32: the lane pairing matches
//     the WMMA B-matrix layout (K0,1 in lanes 0-15; K2,3 in lanes 16-31),
//     A = SH basis replicated across rows (wave-uniform: an aligned 16-sample
//     group is always inside one ray). Exact f32, no extra data movement.
//     Per-sample {tau,tmid,rgb,normal} staged in LDS (4 KB/block).
//   Phase 2 (waves 0-1 after barrier): one wave per ray (block = 2 rays),
//     shfl inclusive scan for the transmittance cumsum over 64 samples,
//     shfl-xor tree reductions for the 8 weighted outputs.
// This removes the 2x16 MB global staging round-trip of the 2-kernel version.
// ---------------------------------------------------------------------------

#define PV_N 8192      // rays
#define PV_S 64        // samples per ray
#define PV_M (PV_N * PV_S)
#define PV_VOXEL 0.02f
#define PV_INV_VOXEL 50.0f

typedef __attribute__((ext_vector_type(2))) float v2f;
typedef __attribute__((ext_vector_type(8))) float v8f;

__device__ __forceinline__ float pv_sigmoid(float x) {
    return 1.0f / (1.0f + expf(-x));
}

__device__ __forceinline__ float pv_incl_scan(float x, int lane) {
#pragma unroll
    for (int off = 1; off < 32; off <<= 1) {
        float t = __shfl_up(x, off, 32);
        if (lane >= off) x += t;
    }
    return x;
}

__global__ __launch_bounds__(256) void pv_fused_kernel(
    const float* __restrict__ rays_o,   // [N,3]
    const float* __restrict__ rays_d,   // [N,3]
    const float* __restrict__ rdnorm,   // [N]
    const float* __restrict__ nearv,    // [N]
    const float* __restrict__ farv,     // [N]
    const float* __restrict__ table,    // [8192, 512, 28]
    const float* __restrict__ beta,     // [1]
    float* __restrict__ out)            // full output buffer
{
    // LDS staging: 128 samples x {tau,tmid,r,g | b,nx,ny,nz}
    __shared__ float4 lds_st[128 * 2];

    const int lane = threadIdx.x & 31;
    const int wave = threadIdx.x >> 5;
    const int n = lane & 15;                  // sample slot within wave
    const bool hi = lane >= 16;               // high half of the lane pair
    const int loc = wave * 16 + n;            // sample within block [0,128)
    const int sid = blockIdx.x * 128 + loc;   // global sample id
    const int ray = sid >> 6;
    const int s = sid & 63;

    // --- sample position along the ray ---
    const float nr = nearv[ray];
    const float fr = farv[ray];
    const float dt = (fr - nr) * (1.0f / (float)PV_S);
    const float tmid = nr + dt * ((float)s + 0.5f);

    const float dx = rays_d[ray * 3 + 0];
    const float dy = rays_d[ray * 3 + 1];
    const float dz = rays_d[ray * 3 + 2];
    const float px = rays_o[ray * 3 + 0] + tmid * dx;
    const float py = rays_o[ray * 3 + 1] + tmid * dy;
    const float pz = rays_o[ray * 3 + 2] + tmid * dz;

    // --- grid coords + trilinear weights ---
    const float gx = px * PV_INV_VOXEL;
    const float gy = py * PV_INV_VOXEL;
    const float gz = pz * PV_INV_VOXEL;
    const float fx = floorf(gx), fy = floorf(gy), fz = floorf(gz);
    const float wx = gx - fx, wy = gy - fy, wz = gz - fz;
    const int ix = (int)fx, iy = (int)fy, iz = (int)fz;

    // --- gather 4 of 8 corners per lane, accumulate emb + grad partials ---
    float e[28];
#pragma unroll
    for (int t = 0; t < 28; ++t) e[t] = 0.0f;
    float grx = 0.0f, gry = 0.0f, grz = 0.0f;

#pragma unroll
    for (int cc = 0; cc < 4; ++cc) {
        const int c = (hi ? 4 : 0) + cc;
        const int oi = (c >> 2) & 1, oj = (c >> 1) & 1, ok = c & 1;
        const int cx = ix + oi, cy = iy + oj, cz = iz + ok;
        // floor-div by 8 via arithmetic shift (matches jnp.floor_divide)
        const int bx = cx >> 3, by = cy >> 3, bz = cz >> 3;
        const int lx = cx & 7, ly = cy & 7, lz = cz & 7;
        unsigned h = (unsigned)bx ^ ((unsigned)by * 2654435761u)
                                  ^ ((unsigned)bz * 805459861u);
        h &= 8191u;                                   // % T_TABLE (pow2)
        const int cell = (lx * 8 + ly) * 8 + lz;
        const float4* p =
            (const float4*)(table + ((long)h * 512 + (long)cell) * 28);

        const float ux = oi ? wx : 1.0f - wx;
        const float uy = oj ? wy : 1.0f - wy;
        const float uz = ok ? wz : 1.0f - wz;
        const float ww = ux * uy * uz;

        const float4 q0 = p[0], q1 = p[1], q2 = p[2], q3 = p[3];
        const float4 q4 = p[4], q5 = p[5], q6 = p[6];

        // analytic d(sdf)/dx partials (sdf = channel 0)
        const float sdfc = q0.x;
        grx += sdfc * (oi ? 1.0f : -1.0f) * uy * uz;
        gry += sdfc * (oj ? 1.0f : -1.0f) * ux * uz;
        grz += sdfc * (ok ? 1.0f : -1.0f) * ux * uy;

        e[0] += ww * q0.x;  e[1] += ww * q0.y;  e[2] += ww * q0.z;  e[3] += ww * q0.w;
        e[4] += ww * q1.x;  e[5] += ww * q1.y;  e[6] += ww * q1.z;  e[7] += ww * q1.w;
        e[8] += ww * q2.x;  e[9] += ww * q2.y;  e[10] += ww * q2.z; e[11] += ww * q2.w;
        e[12] += ww * q3.x; e[13] += ww * q3.y; e[14] += ww * q3.z; e[15] += ww * q3.w;
        e[16] += ww * q4.x; e[17] += ww * q4.y; e[18] += ww * q4.z; e[19] += ww * q4.w;
        e[20] += ww * q5.x; e[21] += ww * q5.y; e[22] += ww * q5.z; e[23] += ww * q5.w;
        e[24] += ww * q6.x; e[25] += ww * q6.y; e[26] += ww * q6.z; e[27] += ww * q6.w;
    }

    // --- pair reduction: lanes n and n+16 both end with full sums ---
#pragma unroll
    for (int t = 0; t < 28; ++t) e[t] += __shfl_xor(e[t], 16, 32);
    grx += __shfl_xor(grx, 16, 32);
    gry += __shfl_xor(gry, 16, 32);
    grz += __shfl_xor(grz, 16, 32);
    grx *= PV_INV_VOXEL; gry *= PV_INV_VOXEL; grz *= PV_INV_VOXEL;

    // --- VolSDF sigma ---
    const float sdf = e[0];
    const float beta_eff = PV_VOXEL + fabsf(beta[0]);
    const float alpha = 1.0f / beta_eff;
    const float sgn = (sdf > 0.0f) ? 1.0f : ((sdf < 0.0f) ? -1.0f : 0.0f);
    const float sigma = alpha * (0.5f + 0.5f * sgn * expm1f(-fabsf(sdf) * alpha));
    const float tau = sigma * dt;

    // --- SH basis (uniform across the wave: one ray per 16-sample group) ---
    float cmp[12];
    cmp[0] = 0.28209479177387814f;
    cmp[1] = 0.4886025119029199f * dy;
    cmp[2] = 0.4886025119029199f * dz;
    cmp[3] = 0.4886025119029199f * dx;
    cmp[4] = 1.0925484305920792f * dx * dy;
    cmp[5] = 1.0925484305920792f * dy * dz;
    cmp[6] = 0.9461746957575601f * dz * dz - 0.31539156525252f;
    cmp[7] = 1.0925484305920792f * dx * dz;
    cmp[8] = 0.5462742152960396f * (dx * dx - dy * dy);
    cmp[9] = 0.0f; cmp[10] = 0.0f; cmp[11] = 0.0f;

    // --- rgb logits via V_WMMA_F32_16X16X4_F32, K=9 padded to 12 ---
    // A[m][k] = cmp[k] (rows replicated), B[k][n] = sh_n[c][k].
    // B layout wants K0,1 in lanes 0-15 and K2,3 in lanes 16-31: exactly our
    // lane pairing, so no data movement is needed.
    float rgbv[3];
#pragma unroll
    for (int c = 0; c < 3; ++c) {
        float shp[12];
#pragma unroll
        for (int l = 0; l < 9; ++l) shp[l] = e[1 + c * 9 + l];
        shp[9] = 0.0f; shp[10] = 0.0f; shp[11] = 0.0f;

        v8f acc = {0.0f, 0.0f, 0.0f, 0.0f, 0.0f, 0.0f, 0.0f, 0.0f};
#pragma unroll
        for (int j = 0; j < 3; ++j) {
            v2f a, b;
            a.x = hi ? cmp[4 * j + 2] : cmp[4 * j + 0];
            a.y = hi ? cmp[4 * j + 3] : cmp[4 * j + 1];
            b.x = hi ? shp[4 * j + 2] : shp[4 * j + 0];
            b.y = hi ? shp[4 * j + 3] : shp[4 * j + 1];
            acc = __builtin_amdgcn_wmma_f32_16x16x4_f32(
                false, a, false, b, (short)0, acc, false, false);
        }
        // every row of column n holds the logit of sample n
        rgbv[c] = pv_sigmoid(acc[0]);
    }

    // --- normals ---
    const float gl = sqrtf(grx * grx + gry * gry + grz * grz);
    const float ginv = 1.0f / fmaxf(gl, 1e-12f);

    // --- stage to LDS + write sdf_grads (divergence only after WMMAs) ---
    if (!hi) {
        lds_st[loc * 2 + 0] = make_float4(tau, tmid, rgbv[0], rgbv[1]);
        out[8 * PV_N + (size_t)sid * 3 + 0] = grx;
        out[8 * PV_N + (size_t)sid * 3 + 1] = gry;
        out[8 * PV_N + (size_t)sid * 3 + 2] = grz;
    } else {
        lds_st[loc * 2 + 1] = make_float4(rgbv[2], grx * ginv, gry * ginv, grz * ginv);
    }

    __syncthreads();

    // --- Phase 2: waves 0-1 render the block's two rays from LDS ---
    if (wave < 2) {
        const int r = blockIdx.x * 2 + wave;
        const int l1 = wave * 64 + lane;          // sample s = lane
        const int l2 = l1 + 32;                   // sample s = lane + 32
        const float4 a0 = lds_st[l1 * 2 + 0];
        const float4 a1 = lds_st[l1 * 2 + 1];
        const float4 b0 = lds_st[l2 * 2 + 0];
        const float4 b1 = lds_st[l2 * 2 + 1];

        const float tau1 = a0.x, tau2 = b0.x;

        // exclusive transmittance: inclusive scan over 64 samples (2/lane)
        float sc1 = pv_incl_scan(tau1, lane);
        const float tot1 = __shfl(sc1, 31, 32);
        float sc2 = pv_incl_scan(tau2, lane) + tot1;
        const float ex1 = sc1 - tau1;
        const float ex2 = sc2 - tau2;
        const float w1 = (1.0f - expf(-tau1)) * expf(-ex1);
        const float w2 = (1.0f - expf(-tau2)) * expf(-ex2);

        float pr[8];
        pr[0] = w1 * a0.z + w2 * b0.z;   // rgb.r
        pr[1] = w1 * a0.w + w2 * b0.w;   // rgb.g
        pr[2] = w1 * a1.x + w2 * b1.x;   // rgb.b
        pr[3] = w1 * a0.y + w2 * b0.y;   // depth (t_mid)
        pr[4] = w1 * a1.y + w2 * b1.y;   // normal.x
        pr[5] = w1 * a1.z + w2 * b1.z;   // normal.y
        pr[6] = w1 * a1.w + w2 * b1.w;   // normal.z
        pr[7] = w1 + w2;                 // acc

#pragma unroll
        for (int q = 0; q < 8; ++q) {
#pragma unroll
            for (int off = 16; off >= 1; off >>= 1)
                pr[q] += __shfl_xor(pr[q], off, 32);
        }

        if (lane == 0) {
            const float inorm = 1.0f / rdnorm[r];
            // output layout (floats): rgb[3N] | depth[N] | normals[3N] |
            //                         acc[N] | sdf_grads[3M] | near[N] | far[N]
            out[r * 3 + 0] = pr[0];
            out[r * 3 + 1] = pr[1];
            out[r * 3 + 2] = pr[2];
            out[3 * PV_N + r] = pr[3] * inorm;
            out[4 * PV_N + r * 3 + 0] = pr[4];
            out[4 * PV_N + r * 3 + 1] = pr[5];
            out[4 * PV_N + r * 3 + 2] = pr[6];
            out[7 * PV_N + r] = pr[7];
            out[8 * PV_N + 3 * PV_M + r] = nearv[r] * inorm;
            out[8 * PV_N + 3 * PV_M + PV_N + r] = farv[r] * inorm;
        }
    }
}

extern "C" void kernel_launch(void* const* d_in, const int* in_sizes, int n_in,
                              void* d_out, int out_size, void* d_ws, size_t ws_size,
                              hipStream_t stream) {
    const float* rays_o = (const float*)d_in[0];
    const float* rays_d = (const float*)d_in[1];
    const float* rdnorm = (const float*)d_in[2];
    const float* nearv  = (const float*)d_in[3];
    const float* farv   = (const float*)d_in[4];
    const float* table  = (const float*)d_in[5];
    const float* beta   = (const float*)d_in[6];

    float* out = (float*)d_out;

    // One block = 8 waves x 16 samples = 128 samples = 2 complete rays.
    const int blocks = PV_M / 128;              // 4096
    pv_fused_kernel<<<blocks, 256, 0, stream>>>(
        rays_o, rays_d, rdnorm, nearv, farv, table, beta, out);
}